// Unsigned_MSELoss_91044716740935
// MI455X (gfx1250) — compile-verified
//
#include <hip/hip_runtime.h>

// Unsigned MSE loss: per-row SSE for +/- sign, segment-sum over sorted batch ids,
// min(pos,neg) per segment, total sum. Pure streaming: ~272MB read, ~0.2 GFLOP
// -> HBM-bound (~12us floor @ 23.3 TB/s). CDNA5 path: async global->LDS DMA
// (ASYNCcnt) with a 4-deep pipeline (Little's law: ~16MB in flight needed to
// saturate 23.3 TB/s @ ~700ns; 3KB/wave * 4096 waves ~= 12MB), wave32 shuffles,
// LDS atomics, NT hints on the read-once streams.

#define MAIN_BLOCKS 512
#define BINS        512
#define ROWS_PER_CHUNK 64   // 64 rows * 64B = 4KB per matrix per chunk
#define TPB         256
#define DEPTH       4       // async pipeline buffers (3 chunks prefetched ahead)

__device__ __forceinline__ void async_load_b128(uint32_t lds_off, uint32_t goff, uint64_t base) {
  // GVS mode: global address = SGPR base + 32-bit VGPR offset; per-lane LDS address from the first VGPR operand
  asm volatile("global_load_async_to_lds_b128 %0, %1, %2 offset:0 th:TH_LOAD_NT"
               :
               : "v"(lds_off), "v"(goff), "s"(base)
               : "memory");
}

__global__ void __launch_bounds__(TPB)
unsigned_mse_main(const float* __restrict__ inp, const float* __restrict__ tgt,
                  const int* __restrict__ batch,
                  float* __restrict__ wsP, float* __restrict__ wsN,
                  int nrows, int nchunks, int cpb)
{
  __shared__ float4 abuf[DEPTH][TPB];   // input  buffers (4 x 4KB)
  __shared__ float4 bbuf[DEPTH][TPB];   // target buffers (4 x 4KB)
  __shared__ float  binsP[BINS];
  __shared__ float  binsN[BINS];

  const int t = threadIdx.x;
  for (int i = t; i < BINS; i += TPB) { binsP[i] = 0.0f; binsN[i] = 0.0f; }
  __syncthreads();

  const int c0 = blockIdx.x * cpb;
  const int c1 = min(c0 + cpb, nchunks);

  if (c0 < c1) {
    const uint64_t inB = (uint64_t)(uintptr_t)inp;
    const uint64_t tgB = (uint64_t)(uintptr_t)tgt;
    // low 32 bits of a generic LDS pointer == wave-relative LDS byte offset
    const uint32_t ldsA = (uint32_t)(uintptr_t)&abuf[0][t];
    const uint32_t ldsB = (uint32_t)(uintptr_t)&bbuf[0][t];
    const uint32_t maxoff = (uint32_t)nrows * 64u - 16u;  // clamp tail (dup reads, harmless)

    // Issue one chunk's loads: 2 async instructions per wave (ASYNCcnt += 2).
    auto issue = [&](int c) {
      uint32_t goff = (uint32_t)c * 4096u + (uint32_t)t * 16u;
      if (goff > maxoff) goff = maxoff;
      const uint32_t boff = ((uint32_t)(c - c0) & (DEPTH - 1)) * (TPB * 16u);
      async_load_b128(ldsA + boff, goff, inB);
      async_load_b128(ldsB + boff, goff, tgB);
    };

    // Prologue: prefetch up to 3 chunks ahead.
    #pragma unroll
    for (int k = 0; k < DEPTH - 1; ++k)
      if (c0 + k < c1) issue(c0 + k);

    for (int c = c0; c < c1; ++c) {
      if (c + (DEPTH - 1) < c1) issue(c + (DEPTH - 1));

      // Wait until chunk c's 2 loads are done: allowed outstanding = 2 * chunks ahead.
      // (async loads complete in order; immediates only, so constant branches)
      int ahead = c1 - 1 - c;
      if (ahead > DEPTH - 1) ahead = DEPTH - 1;
      if      (ahead == 3) asm volatile("s_wait_asynccnt 0x6" ::: "memory");
      else if (ahead == 2) asm volatile("s_wait_asynccnt 0x4" ::: "memory");
      else if (ahead == 1) asm volatile("s_wait_asynccnt 0x2" ::: "memory");
      else                 asm volatile("s_wait_asynccnt 0x0" ::: "memory");

      // Thread t owns float4 #t of the chunk: row = t/4, feature quarter = t%4.
      // LDS reads are linear across lanes -> conflict free. Each wave only reads
      // bytes its own async ops loaded -> no block barrier needed in this loop.
      const int buf = (c - c0) & (DEPTH - 1);
      const float4 a = abuf[buf][t];
      const float4 b = bbuf[buf][t];
      const float dx = a.x - b.x, dy = a.y - b.y, dz = a.z - b.z, dw = a.w - b.w;
      const float sx = a.x + b.x, sy = a.y + b.y, sz = a.z + b.z, sw = a.w + b.w;
      float sp = dx * dx + dy * dy + dz * dz + dw * dw;
      float sn = sx * sx + sy * sy + sz * sz + sw * sw;
      // Reduce the 4 feature-quarters of a row (lanes 4k..4k+3) within the wave32.
      sp += __shfl_xor(sp, 1, 32);
      sp += __shfl_xor(sp, 2, 32);
      sn += __shfl_xor(sn, 1, 32);
      sn += __shfl_xor(sn, 2, 32);
      if ((t & 3) == 0) {
        const int row = c * ROWS_PER_CHUNK + (t >> 2);
        if (row < nrows) {
          const int bin = __builtin_nontemporal_load(&batch[row]);  // sorted, read-once
          atomicAdd(&binsP[bin], sp);          // ds_add_f32, low contention spread
          atomicAdd(&binsN[bin], sn);
        }
      }
    }
  }

  __syncthreads();
  // Full per-block slice written (including zeros) -> d_ws fully initialized.
  float* wp = wsP + (size_t)blockIdx.x * BINS;
  float* wn = wsN + (size_t)blockIdx.x * BINS;
  for (int i = t; i < BINS; i += TPB) { wp[i] = binsP[i]; wn[i] = binsN[i]; }
}

__global__ void __launch_bounds__(512)
unsigned_mse_final(const float* __restrict__ wsP, const float* __restrict__ wsN,
                   float* __restrict__ out)
{
  const int tid = threadIdx.x;   // tid == bin id (0..511)
  float p = 0.0f, n = 0.0f;
  for (int blk = 0; blk < MAIN_BLOCKS; ++blk) {      // fixed order -> deterministic
    p += wsP[(size_t)blk * BINS + tid];              // coalesced across tid, L2-resident
    n += wsN[(size_t)blk * BINS + tid];
  }
  float v = fminf(p, n);   // untouched bins are exactly 0 -> contribute 0

  // Block reduction of 512 lanes: wave32 shuffle tree + LDS combine.
  #pragma unroll
  for (int o = 1; o < 32; o <<= 1) v += __shfl_xor(v, o, 32);
  __shared__ float part[16];
  if ((tid & 31) == 0) part[tid >> 5] = v;
  __syncthreads();
  if (tid < 32) {
    float s = (tid < 16) ? part[tid] : 0.0f;
    #pragma unroll
    for (int o = 1; o < 16; o <<= 1) s += __shfl_xor(s, o, 32);
    if (tid == 0) out[0] = s;
  }
}

extern "C" void kernel_launch(void* const* d_in, const int* in_sizes, int n_in,
                              void* d_out, int out_size, void* d_ws, size_t ws_size,
                              hipStream_t stream) {
  const float* inp   = (const float*)d_in[0];   // [N,16] f32
  const float* tgt   = (const float*)d_in[1];   // [N,16] f32
  const int*   batch = (const int*)d_in[2];     // [N] sorted segment ids
  // d_in[3] = batch_size (unused: unused bins stay 0 and min(0,0)=0)

  const int nrows   = in_sizes[0] / 16;
  const int nchunks = (nrows + ROWS_PER_CHUNK - 1) / ROWS_PER_CHUNK;
  const int cpb     = (nchunks + MAIN_BLOCKS - 1) / MAIN_BLOCKS;

  float* wsP = (float*)d_ws;                          // [MAIN_BLOCKS][BINS]
  float* wsN = wsP + (size_t)MAIN_BLOCKS * BINS;      // [MAIN_BLOCKS][BINS]  (2MB total)

  unsigned_mse_main<<<MAIN_BLOCKS, TPB, 0, stream>>>(inp, tgt, batch, wsP, wsN,
                                                     nrows, nchunks, cpb);
  unsigned_mse_final<<<1, 512, 0, stream>>>(wsP, wsN, (float*)d_out);
}